// TrueReservoir_48636209660444
// MI455X (gfx1250) — compile-verified
//
#include <hip/hip_runtime.h>
#include <math.h>

// MI455X / gfx1250 reservoir (echo-state) kernel.
//   Phase 0: init workspace (barrier counters + state ping buffer)    [tiny]
//   Phase 1: u = x @ W_in^T  via V_WMMA_F32_16X16X4_F32 -> d_out      [bulk GEMM]
//   Phase 2: persistent cooperative scan over S=2048 steps:
//              s_t = 0.95*tanh(u_t + s_{t-1} @ W_res^T) + 0.05*s_{t-1}
//            64 single-wave WGs, W_res column-slice in LDS, fp32 WMMA,
//            agent-scope atomic barrier per step.

typedef __attribute__((ext_vector_type(2))) float v2f;
typedef __attribute__((ext_vector_type(8))) float v8f;

#define LEAK 0.95f
#define B_ 16
#define S_ 2048
#define I_ 512
#define H_ 1024
#define NWG 64   // phase-2 workgroups: one 16-column tile each

// ---------------------------------------------------------------- init
__global__ __launch_bounds__(256) void rsv_init(const float* __restrict__ init_state,
                                                float* __restrict__ state0,
                                                unsigned* __restrict__ sync) {
    if (blockIdx.x == 0 && threadIdx.x == 0) { sync[0] = 0u; sync[1] = 0u; }
    int i = blockIdx.x * blockDim.x + threadIdx.x;
    int n = B_ * H_;
    for (; i < n; i += gridDim.x * blockDim.x) state0[i] = init_state[i];
}

// ---------------------------------------------------------------- phase 1
// C[flat_m, n] = sum_k x[flat_m, k] * W_in[n, k];  flat_m = b*S + s (32768 rows)
// One wave = one 16-row M tile x four 16-col N tiles (A fragment reused 4x).
__global__ __launch_bounds__(256) void rsv_inproj(const float* __restrict__ x,
                                                  const float* __restrict__ Win,
                                                  float* __restrict__ out) {
    const int lane   = threadIdx.x & 31;
    const int wid    = blockIdx.x * (blockDim.x >> 5) + (threadIdx.x >> 5);
    const int m_tile = wid >> 4;            // 0..2047
    const int n_grp  = wid & 15;            // 0..15  (64 columns each)
    const int mrow   = lane & 15;
    const int hi     = lane >> 4;           // 0: K=0,1  1: K=2,3 (A); M+8 (C/D)
    const int klo    = hi * 2;
    const int row    = m_tile * 16 + mrow;  // flat (b*S+s)

    const float* xrow = x + (size_t)row * I_ + klo;
    v8f acc[4] = {};
#pragma unroll 4
    for (int k0 = 0; k0 < I_; k0 += 4) {
        v2f a = *(const v2f*)(xrow + k0);
#pragma unroll
        for (int j = 0; j < 4; ++j) {
            const int n = n_grp * 64 + j * 16 + mrow;     // B[k][n] = W_in[n][k]
            v2f b = *(const v2f*)(Win + (size_t)n * I_ + k0 + klo);
            acc[j] = __builtin_amdgcn_wmma_f32_16x16x4_f32(
                false, a, false, b, (short)0, acc[j], false, false);
        }
    }
#pragma unroll
    for (int j = 0; j < 4; ++j) {
        const int n = n_grp * 64 + j * 16 + mrow;
#pragma unroll
        for (int r = 0; r < 8; ++r) {
            const int m = m_tile * 16 + r + hi * 8;       // C/D: VGPR r -> M=r / M=r+8
            out[(size_t)m * H_ + n] = acc[j][r];
        }
    }
}

// ---------------------------------------------------------------- barrier
__device__ inline void grid_barrier(unsigned* __restrict__ sync, unsigned gen) {
    __threadfence();   // make this wave's state/out stores agent-visible
    if ((threadIdx.x & 31) == 0) {
        unsigned old = __hip_atomic_fetch_add(&sync[0], 1u, __ATOMIC_ACQ_REL,
                                              __HIP_MEMORY_SCOPE_AGENT);
        if (old == NWG - 1u) {
            __hip_atomic_store(&sync[0], 0u, __ATOMIC_RELAXED, __HIP_MEMORY_SCOPE_AGENT);
            __hip_atomic_store(&sync[1], gen, __ATOMIC_RELEASE, __HIP_MEMORY_SCOPE_AGENT);
        }
    }
    while (__hip_atomic_load(&sync[1], __ATOMIC_ACQUIRE, __HIP_MEMORY_SCOPE_AGENT) < gen)
        __builtin_amdgcn_s_sleep(1);
}

// ---------------------------------------------------------------- phase 2
// Persistent scan. WG w owns output columns [16w, 16w+16).
// LDS holds B[k][n] = W_res[nbase+n][k] as K-pair-interleaved v2f:
//   ldsB[(k>>1)*16 + n] = { B[k][n], B[k+1][n] }   (k even)
// -> per-WMMA B fragment is one conflict-free 8B LDS load per lane.
__global__ __launch_bounds__(32) void rsv_scan(const float* __restrict__ Wres,
                                               float* __restrict__ out,  // u in, states out
                                               float* __restrict__ st0,
                                               float* __restrict__ st1,
                                               unsigned* __restrict__ sync) {
    extern __shared__ v2f ldsB[];           // 512*16 v2f = 64 KB
    const int lane  = threadIdx.x;          // single wave per block
    const int nbase = blockIdx.x * 16;
    const int mrow  = lane & 15;
    const int hi    = lane >> 4;
    const int klo   = hi * 2;

    // Load W_res column slice into LDS (coalesced 8B global reads).
    for (int i = lane; i < 16 * (H_ / 2); i += 32) {
        const int n = i >> 9;               // 0..15
        const int p = i & 511;              // k pair index
        ldsB[p * 16 + n] = *(const v2f*)(Wres + (size_t)(nbase + n) * H_ + 2 * p);
    }
    __syncthreads();

    float* prev = st0;
    float* next = st1;
    unsigned gen = 0;

    for (int t = 0; t < S_; ++t) {
        const float* arow = prev + mrow * H_ + klo;
        v8f acc[4] = {};                    // 4 accumulators break the C RAW chain
#pragma unroll 2
        for (int k0 = 0; k0 < H_; k0 += 16) {
#pragma unroll
            for (int j = 0; j < 4; ++j) {
                const int k = k0 + 4 * j;
                v2f a = *(const v2f*)(arow + k);
                v2f b = ldsB[((k >> 1) + hi) * 16 + mrow];
                acc[j] = __builtin_amdgcn_wmma_f32_16x16x4_f32(
                    false, a, false, b, (short)0, acc[j], false, false);
            }
        }
        v8f c = (acc[0] + acc[1]) + (acc[2] + acc[3]);

        const int n = nbase + mrow;
#pragma unroll
        for (int r = 0; r < 8; ++r) {
            const int m = r + hi * 8;
            const size_t oidx = ((size_t)m * S_ + t) * H_ + n;
            const float u  = out[oidx];                  // precomputed x@W_in^T
            const float pv = prev[m * H_ + n];
            const float nv = LEAK * tanhf(u + c[r]) + (1.0f - LEAK) * pv;
            next[m * H_ + n] = nv;
            out[oidx]        = nv;
        }

        ++gen;
        grid_barrier(sync, gen);
        float* tp = prev; prev = next; next = tp;
    }
}

// ---------------------------------------------------------------- launch
extern "C" void kernel_launch(void* const* d_in, const int* in_sizes, int n_in,
                              void* d_out, int out_size, void* d_ws, size_t ws_size,
                              hipStream_t stream) {
    const float* x    = (const float*)d_in[0];   // [B,S,I]
    const float* Win  = (const float*)d_in[1];   // [H,I]
    const float* Wres = (const float*)d_in[2];   // [H,H]
    const float* init = (const float*)d_in[3];   // [B,H]
    float* out = (float*)d_out;                  // [B,S,H]

    char* ws = (char*)d_ws;
    float*    st0  = (float*)ws;                       // 64 KB
    float*    st1  = (float*)(ws + 64 * 1024);         // 64 KB
    unsigned* sync = (unsigned*)(ws + 128 * 1024);     // counter, phase

    rsv_init<<<64, 256, 0, stream>>>(init, st0, sync);
    // 2048 M-tiles * 16 N-groups = 32768 waves / 8 waves per block
    rsv_inproj<<<4096, 256, 0, stream>>>(x, Win, out);
    rsv_scan<<<NWG, 32, 64 * 1024, stream>>>(Wres, out, st0, st1, sync);
}